// Distance_38817914422096
// MI455X (gfx1250) — compile-verified
//
#include <hip/hip_runtime.h>
#include <hip/hip_bf16.h>

// Pairwise Euclidean distance, strict lower triangle, via WMMA Gram matrix:
//   d(i,j) = sqrt(max(n_i + n_j - 2 * <x_i, x_j>, EPS))
// Gram computed with V_WMMA_F32_16X16X4_F32 (native fp32 matrix op), one wave
// per 32x32 macro-tile = 2x2 register tile of 16x16 WMMA tiles:
// per K-step 4 global_load_b64 feed 4 WMMAs (1 load : 1 wmma).

typedef __attribute__((ext_vector_type(2))) float v2f;
typedef __attribute__((ext_vector_type(4))) float v4f;
typedef __attribute__((ext_vector_type(8))) float v8f;

#define BATCH 8
#define SEQ   512
#define DIM   256
#define TRI   (SEQ * (SEQ - 1) / 2)        // 130816
#define MT    32                           // macro-tile edge
#define NMT   (SEQ / MT)                   // 16 macro-tiles per edge
#define NPAIR (NMT * (NMT + 1) / 2)        // 136 lower-tri macro pairs
#define EPSQ  1e-7f

// ---------------------------------------------------------------------------
// Kernel 1: squared L2 norm of each row. One wave per row, shuffle-reduce.
// ---------------------------------------------------------------------------
__global__ __launch_bounds__(256) void row_norms_kernel(
    const float* __restrict__ x, float* __restrict__ norms) {
  const int wave = (int)((blockIdx.x * blockDim.x + threadIdx.x) >> 5);
  const int lane = (int)(threadIdx.x & 31);
  const v4f* p = (const v4f*)(x + (size_t)wave * DIM) + lane * 2;
  v4f a = p[0];
  v4f b = p[1];
  float s = a.x * a.x + a.y * a.y + a.z * a.z + a.w * a.w +
            b.x * b.x + b.y * b.y + b.z * b.z + b.w * b.w;
#pragma unroll
  for (int off = 16; off > 0; off >>= 1) s += __shfl_xor(s, off, 32);
  if (lane == 0) norms[wave] = s;
}

// ---------------------------------------------------------------------------
// Distance epilogue for one 16x16 Gram tile held in a WMMA f32 accumulator.
// C/D layout (wave32): lane l, VGPR v -> N = l&15, M = v + 8*(l>>4).
// Lanes 0..15 of each v share i and have consecutive j -> 64B stores.
// ---------------------------------------------------------------------------
__device__ __forceinline__ void epilogue_tile(
    v8f c, int row_base, int col_base, int half, int l16,
    const float* __restrict__ norms_b, float* __restrict__ outb) {
  const float ncol = norms_b[col_base + l16];
  const int   ibase = row_base + half * 8;
  const int   j = col_base + l16;
#pragma unroll
  for (int v = 0; v < 8; ++v) {
    const int i = ibase + v;
    if (i > j) {
      float d2 = norms_b[i] + ncol - 2.0f * c[v];
      d2 = fmaxf(d2, EPSQ);  // clamp >= 1e-7 -> raw v_sqrt_f32 is safe
      outb[i * (i - 1) / 2 + j] = __builtin_amdgcn_sqrtf(d2);
    }
  }
}

// ---------------------------------------------------------------------------
// Kernel 2: one wave per 32x32 macro-tile (lower triangle incl. diagonal).
// Fragment layouts (wave32, f32, K=4):
//   A 16x4 : lane l, VGPR v -> M = l&15, K = 2*(l>>4) + v   (float2/lane)
//   B 4x16 : lane l, VGPR v -> N = l&15, K = 2*(l>>4) + v   (float2/lane)
// Both are contiguous 8B-aligned float2 loads from row-major X.
// ---------------------------------------------------------------------------
__global__ __launch_bounds__(256) void dist_tile_kernel(
    const float* __restrict__ x, const float* __restrict__ norms,
    float* __restrict__ out) {
  const int lane = (int)(threadIdx.x & 31);
  const int wave = (int)(threadIdx.x >> 5);
  const int t    = (int)blockIdx.x * 8 + wave;  // macro pair id, 0..NPAIR-1
  const int b    = (int)blockIdx.y;             // batch

  // Decode t -> (ti, tj): t = ti*(ti+1)/2 + tj, ti >= tj. Wave-uniform.
  int ti = (int)((sqrtf(8.0f * (float)t + 1.0f) - 1.0f) * 0.5f);
  while ((ti + 1) * (ti + 2) / 2 <= t) ++ti;
  while (ti * (ti + 1) / 2 > t) --ti;
  const int tj = t - ti * (ti + 1) / 2;

  const int row0 = ti * MT;      // macro rows i (larger index side)
  const int col0 = tj * MT;      // macro cols j
  const int half = lane >> 4;    // 0 or 1
  const int l16  = lane & 15;

  const float* xb = x + (size_t)b * SEQ * DIM;
  const float* ar0 = xb + (size_t)(row0 + l16) * DIM + 2 * half;
  const float* ar1 = ar0 + (size_t)16 * DIM;
  const float* bc0 = xb + (size_t)(col0 + l16) * DIM + 2 * half;
  const float* bc1 = bc0 + (size_t)16 * DIM;

  v8f c00 = {0.f, 0.f, 0.f, 0.f, 0.f, 0.f, 0.f, 0.f};
  v8f c01 = c00, c10 = c00, c11 = c00;

#pragma unroll 4
  for (int k = 0; k < DIM; k += 4) {
    v2f a0 = *(const v2f*)(ar0 + k);   // global_load_b64 x4
    v2f a1 = *(const v2f*)(ar1 + k);
    v2f b0 = *(const v2f*)(bc0 + k);
    v2f b1 = *(const v2f*)(bc1 + k);
    // 8 args: (neg_a, A, neg_b, B, c_mod, C, reuse_a, reuse_b)
    c00 = __builtin_amdgcn_wmma_f32_16x16x4_f32(false, a0, false, b0, (short)0, c00, false, false);
    c01 = __builtin_amdgcn_wmma_f32_16x16x4_f32(false, a0, false, b1, (short)0, c01, false, false);
    c10 = __builtin_amdgcn_wmma_f32_16x16x4_f32(false, a1, false, b0, (short)0, c10, false, false);
    c11 = __builtin_amdgcn_wmma_f32_16x16x4_f32(false, a1, false, b1, (short)0, c11, false, false);
  }

  const float* norms_b = norms + b * SEQ;
  float* outb = out + (size_t)b * TRI;
  epilogue_tile(c00, row0,      col0,      half, l16, norms_b, outb);
  epilogue_tile(c01, row0,      col0 + 16, half, l16, norms_b, outb);
  epilogue_tile(c10, row0 + 16, col0,      half, l16, norms_b, outb);
  epilogue_tile(c11, row0 + 16, col0 + 16, half, l16, norms_b, outb);
}

extern "C" void kernel_launch(void* const* d_in, const int* in_sizes, int n_in,
                              void* d_out, int out_size, void* d_ws, size_t ws_size,
                              hipStream_t stream) {
  const float* x = (const float*)d_in[0];
  float* out = (float*)d_out;
  float* norms = (float*)d_ws;  // BATCH*SEQ floats = 16 KB scratch

  // 4096 rows, one wave each, 8 waves per 256-thread block -> 512 blocks.
  row_norms_kernel<<<dim3((BATCH * SEQ) / 8), 256, 0, stream>>>(x, norms);

  // 136 macro pairs per batch, 8 waves per block -> grid (17, 8).
  dist_tile_kernel<<<dim3(NPAIR / 8, BATCH), 256, 0, stream>>>(x, norms, out);
}